// QCGraph_43662637531494
// MI455X (gfx1250) — compile-verified
//
#include <hip/hip_runtime.h>
#include <math.h>

typedef float v2f __attribute__((ext_vector_type(2)));
typedef float v8f __attribute__((ext_vector_type(8)));

#define N_AG      128
#define STATE_D   256
#define EMBED     64
#define DIN       160   // RNN_H + N_ACTIONS
#define KSTEPS    40    // DIN / 4
#define BATCH     1024
#define XS        164   // padded LDS stride for 160-wide rows
#define SCS       132   // padded LDS stride for 128-wide score rows
#define HSTR      68    // padded LDS stride for 64-wide hidden rows

// float offsets into the dynamic-LDS arena
#define OFF_X     0                       // x = [hidden|actions], 128 x XS
#define OFF_WAT   (OFF_X   + N_AG*XS)     // Wa^T [n][k]; later aliased by agent_all
#define OFF_WGT   (OFF_WAT + N_AG*XS)     // Wg^T [n][k], 64 x XS
#define OFF_SC    (OFF_WGT + EMBED*XS)    // agent_score 128 x SCS; later aliased by hidden
#define OFF_DD    (OFF_SC  + N_AG*SCS)    // d[i] = sqrt(colsum)
#define OFF_ZV    (OFF_DD  + N_AG)        // z = hidden @ wf
#define OFF_B1    (OFF_ZV  + N_AG)        // b1 = states @ hyper_b1_w + b
#define OFF_WF    (OFF_B1  + EMBED)       // wf = states @ hyper_wf_w + b
#define OFF_U64   (OFF_WF  + EMBED)       // relu(states @ v_w1 + v_b1)
#define OFF_RED   (OFF_U64 + EMBED)       // block-reduction scratch
#define OFF_VS    (OFF_RED + 8)           // v scalar
#define OFF_CNT   (OFF_VS  + 4)           // int colcount[128]
#define OFF_JJ    (OFF_CNT + N_AG)        // int argmax j[128]
#define LDS_FLOATS (OFF_JJ + N_AG)

__launch_bounds__(256, 2)   // truthful: LDS limits us to 1 WG/WGP = 2 waves/SIMD
__global__ void qcgraph_kernel(
    const float* __restrict__ act,  const float* __restrict__ st_g,
    const float* __restrict__ hs,   const float* __restrict__ gu,
    const float* __restrict__ waw,  const float* __restrict__ wab,
    const float* __restrict__ gw1w, const float* __restrict__ gw1b,
    const float* __restrict__ hb1w, const float* __restrict__ hb1b,
    const float* __restrict__ hwfw, const float* __restrict__ hwfb,
    const float* __restrict__ vw1,  const float* __restrict__ vb1,
    const float* __restrict__ vw2,  const float* __restrict__ vb2,
    float* __restrict__ out)
{
    extern __shared__ float lds[];
    float* X   = lds + OFF_X;
    float* WAT = lds + OFF_WAT;  float* AA = WAT;   // aliased after GEMM1
    float* WGT = lds + OFF_WGT;
    float* SC  = lds + OFF_SC;   float* H  = SC;    // aliased after argmax
    float* DD  = lds + OFF_DD;
    float* ZV  = lds + OFF_ZV;
    float* B1  = lds + OFF_B1;
    float* WFv = lds + OFF_WF;
    float* U6  = lds + OFF_U64;
    float* RED = lds + OFF_RED;
    float* VS  = lds + OFF_VS;
    int*   CNT = (int*)(lds + OFF_CNT);
    int*   JJ  = (int*)(lds + OFF_JJ);

    const int tid  = threadIdx.x;
    const int b    = blockIdx.x;
    const int lane = tid & 31;
    const int wid  = tid >> 5;

    // ---------------- Phase 1: stage into LDS + per-batch state vectors ----
    const float* hsb = hs + (size_t)b * (N_AG * 128);
    for (int q = tid; q < N_AG * 32; q += 256) {           // hidden -> X[:,0:128]
        int i = q >> 5, c = (q & 31) << 2;
        *(float4*)&X[i * XS + c] = *(const float4*)&hsb[i * 128 + c];
    }
    const float* actb = act + (size_t)b * (N_AG * 32);
    for (int q = tid; q < N_AG * 8; q += 256) {            // actions -> X[:,128:160]
        int i = q >> 3, c = (q & 7) << 2;
        *(float4*)&X[i * XS + 128 + c] = *(const float4*)&actb[i * 32 + c];
    }
    for (int s = tid; s < DIN * N_AG; s += 256) {          // Wa (160x128) -> Wa^T
        int k = s >> 7, n = s & 127;
        WAT[n * XS + k] = waw[s];
    }
    for (int s = tid; s < DIN * EMBED; s += 256) {         // Wg (160x64) -> Wg^T
        int k = s >> 6, n = s & 63;
        WGT[n * XS + k] = gw1w[s];
    }
    if (tid < N_AG) CNT[tid] = 1;                          // diagonal of cluster
    const float* stb = st_g + (size_t)b * STATE_D;
    if (tid < EMBED) {                                     // b1 = st @ hyper_b1_w + b
        float a0 = hb1b[tid];
        for (int k = 0; k < STATE_D; ++k) a0 += stb[k] * hb1w[k * EMBED + tid];
        B1[tid] = a0;
    } else if (tid < 2 * EMBED) {                          // wf = st @ hyper_wf_w + b
        int t = tid - EMBED;
        float a0 = hwfb[t];
        for (int k = 0; k < STATE_D; ++k) a0 += stb[k] * hwfw[k * EMBED + t];
        WFv[t] = a0;
    } else if (tid < 3 * EMBED) {                          // relu(st @ v_w1 + v_b1)
        int t = tid - 2 * EMBED;
        float a0 = vb1[t];
        for (int k = 0; k < STATE_D; ++k) a0 += stb[k] * vw1[k * EMBED + t];
        U6[t] = fmaxf(a0, 0.0f);
    }
    __syncthreads();

    // ----- Phase 2: GEMM1 score = x @ Wa  (f32 WMMA, 8 fused N-tiles) ------
    const int row  = lane & 15;            // M row (A) / N col (B) within tile
    const int koff = (lane >> 4) << 1;     // K sub-offset per half-wave
    const int m0   = wid * 16;             // wave owns one M-tile row
    const int mb   = m0 + ((lane >> 4) << 3);

    {
        const float* Ap = &X[(m0 + row) * XS + koff];
        const float* Bp = &WAT[row * XS + koff];
        v8f acc[8] = {};
        #pragma unroll
        for (int kk = 0; kk < KSTEPS; ++kk) {
            v2f a = *(const v2f*)(Ap + 4 * kk);            // 1 A-pair, shared by 8 tiles
            #pragma unroll
            for (int t = 0; t < 8; ++t) {
                v2f bb = *(const v2f*)(Bp + t * 16 * XS + 4 * kk);
                acc[t] = __builtin_amdgcn_wmma_f32_16x16x4_f32(
                             false, a, false, bb, (short)0, acc[t], false, false);
            }
        }
        #pragma unroll
        for (int t = 0; t < 8; ++t) {
            const int n = t * 16 + row;
            const float bias = wab[n];
            #pragma unroll
            for (int v = 0; v < 8; ++v) SC[(mb + v) * SCS + n] = acc[t][v] + bias;
        }
    }
    __syncthreads();

    // ---------------- Phase 3: gumbel + per-row argmax -> sparse cluster ----
    const float* gub = gu + (size_t)b * (N_AG * N_AG);
    for (int r = 0; r < 16; ++r) {
        const int i = m0 + r;
        float best = -INFINITY; int bidx = N_AG;
        for (int c = lane; c < N_AG; c += 32) {
            float sc = SC[i * SCS + c];
            if (c == i) sc = -3.0e38f;                      // eye * (-1e16) mask
            float u = gub[i * N_AG + c] * (1.0f - 2e-6f) + 1e-6f;
            float g = -logf(-logf(u));
            float t = sc + g;
            if (t > best || (t == best && c < bidx)) { best = t; bidx = c; }
        }
        #pragma unroll
        for (int m = 16; m >= 1; m >>= 1) {
            float ob = __shfl_xor(best, m, 32);
            int   oi = __shfl_xor(bidx, m, 32);
            if (ob > best || (ob == best && oi < bidx)) { best = ob; bidx = oi; }
        }
        if (lane == 0) { JJ[i] = bidx; atomicAdd(&CNT[bidx], 1); }
    }
    __syncthreads();
    if (tid < N_AG) DD[tid] = sqrtf((float)CNT[tid]);       // d = sqrt(colsum)
    __syncthreads();

    // ---------------- Phase 4: agent_all = C' @ x  (sparse: 2 rows/row) ----
    {
        const int i = tid >> 1;
        const int kb = (tid & 1) * 80;
        const int j = JJ[i];
        const float di = DD[i], dj = DD[j];
        for (int c = 0; c < 80; c += 4) {
            float4 xi = *(const float4*)&X[i * XS + kb + c];
            float4 xj = *(const float4*)&X[j * XS + kb + c];
            float4 r;
            r.x = di * (di * xi.x + dj * xj.x);
            r.y = di * (di * xi.y + dj * xj.y);
            r.z = di * (di * xi.z + dj * xj.z);
            r.w = di * (di * xi.w + dj * xj.w);
            *(float4*)&AA[i * XS + kb + c] = r;
        }
    }
    __syncthreads();

    // ----- Phase 5: GEMM2 hidden = elu(agent_all @ Wg + b), 4 fused tiles --
    {
        const float* Ap = &AA[(m0 + row) * XS + koff];
        const float* Bp = &WGT[row * XS + koff];
        v8f acc[4] = {};
        #pragma unroll
        for (int kk = 0; kk < KSTEPS; ++kk) {
            v2f a = *(const v2f*)(Ap + 4 * kk);
            #pragma unroll
            for (int t = 0; t < 4; ++t) {
                v2f bb = *(const v2f*)(Bp + t * 16 * XS + 4 * kk);
                acc[t] = __builtin_amdgcn_wmma_f32_16x16x4_f32(
                             false, a, false, bb, (short)0, acc[t], false, false);
            }
        }
        #pragma unroll
        for (int t = 0; t < 4; ++t) {
            const int n = t * 16 + row;
            const float bias = gw1b[n] + B1[n];
            #pragma unroll
            for (int v = 0; v < 8; ++v) {
                float h = acc[t][v] + bias;
                H[(mb + v) * HSTR + n] = (h > 0.0f) ? h : expm1f(h);   // elu
            }
        }
    }
    __syncthreads();

    // ---------------- Phase 6: z = hidden @ wf ; v scalar ------------------
    for (int r = 0; r < 16; ++r) {
        const int i = m0 + r;
        float p = H[i * HSTR + lane] * WFv[lane]
                + H[i * HSTR + lane + 32] * WFv[lane + 32];
        #pragma unroll
        for (int m = 16; m >= 1; m >>= 1) p += __shfl_xor(p, m, 32);
        if (lane == 0) ZV[i] = p;
    }
    if (wid == 0) {
        float p = U6[lane] * vw2[lane] + U6[lane + 32] * vw2[lane + 32];
        #pragma unroll
        for (int m = 16; m >= 1; m >>= 1) p += __shfl_xor(p, m, 32);
        if (lane == 0) { float v = p + vb2[0]; VS[0] = v; out[BATCH + b] = v; }
    }
    __syncthreads();

    // ------- Phase 7: pooled = sum_i (z_i + v) * (C' z)_i  ------------------
    if (tid < N_AG) {
        const int i = tid, j = JJ[i];
        const float di = DD[i], dj = DD[j];
        const float cz = di * (di * ZV[i] + dj * ZV[j]);
        float term = (ZV[i] + VS[0]) * cz;
        #pragma unroll
        for (int m = 16; m >= 1; m >>= 1) term += __shfl_xor(term, m, 32);
        if (lane == 0) RED[wid] = term;
    }
    __syncthreads();
    if (tid == 0) out[b] = RED[0] + RED[1] + RED[2] + RED[3];
}

extern "C" void kernel_launch(void* const* d_in, const int* in_sizes, int n_in,
                              void* d_out, int out_size, void* d_ws, size_t ws_size,
                              hipStream_t stream) {
    (void)in_sizes; (void)n_in; (void)d_ws; (void)ws_size; (void)out_size;
    const float* act  = (const float*)d_in[1];
    const float* stt  = (const float*)d_in[2];
    const float* hs   = (const float*)d_in[3];
    const float* gu   = (const float*)d_in[4];
    const float* waw  = (const float*)d_in[5];
    const float* wab  = (const float*)d_in[6];
    const float* gw1w = (const float*)d_in[9];
    const float* gw1b = (const float*)d_in[10];
    const float* hb1w = (const float*)d_in[11];
    const float* hb1b = (const float*)d_in[12];
    const float* hwfw = (const float*)d_in[13];
    const float* hwfb = (const float*)d_in[14];
    const float* vw1  = (const float*)d_in[15];
    const float* vb1  = (const float*)d_in[16];
    const float* vw2  = (const float*)d_in[17];
    const float* vb2  = (const float*)d_in[18];
    float* out = (float*)d_out;

    const size_t shmem = (size_t)LDS_FLOATS * sizeof(float);
    qcgraph_kernel<<<dim3(BATCH), dim3(256), shmem, stream>>>(
        act, stt, hs, gu, waw, wab, gw1w, gw1b, hb1w, hb1b,
        hwfw, hwfb, vw1, vb1, vw2, vb2, out);
}